// SelfConv2d_89739046682626
// MI455X (gfx1250) — compile-verified
//
#include <hip/hip_runtime.h>

// Problem constants (from reference)
#define Bn 16
#define Cn 64
#define Hn 224
#define Wn 224
#define On 128
#define PHW 16
#define OHn 222   // H - 3 + 1
#define OWn 222   // W - 3 + 1
#define NXB 14    // ceil(222/16)
#define HWn (Hn * Wn)
#define OHWn (OHn * OWn)

typedef __attribute__((ext_vector_type(2))) float v2f;
typedef __attribute__((ext_vector_type(4))) float v4f;
typedef __attribute__((ext_vector_type(8))) float v8f;

// ---------------------------------------------------------------------------
// Kernel 0: pbsum[p] = sum_c patch_bias[c, p]   (p = ph*16+pw, 256 entries)
// ---------------------------------------------------------------------------
__global__ __launch_bounds__(256) void k_bias_sum(const float* __restrict__ pb,
                                                  float* __restrict__ pbsum) {
    int i = threadIdx.x;          // 0..255
    float acc = 0.0f;
#pragma unroll
    for (int c = 0; c < Cn; ++c)
        acc += pb[c * 256 + i];
    pbsum[i] = acc;
}

// ---------------------------------------------------------------------------
// Kernel 1: channel reduction, vectorized 4 pixels/thread (b128 path).
//   S[b,h,w] = pbsum[h%16,w%16] + sum_c x[b,c,h,w] * pw[c,h%16,w%16]
// W = 224 is a multiple of 4 and the 4-pixel group never crosses a 16-wide
// patch row, so x / pw / pbsum / S are all naturally float4-aligned.
// ---------------------------------------------------------------------------
__global__ __launch_bounds__(256) void k_reduce_c(const float* __restrict__ x,
                                                  const float* __restrict__ pw,
                                                  const float* __restrict__ pbsum,
                                                  float* __restrict__ S) {
    int t    = blockIdx.x * 256 + threadIdx.x;    // over B*H*W/4 = 200704 (exact)
    int base = t * 4;
    int w4   = base % Wn;                          // multiple of 4
    int hw   = base % HWn;
    int h    = hw / Wn;
    int b    = base / HWn;
    int phw  = (h & 15) * PHW + (w4 & 15);         // multiple of 4 within row

    const v4f* xp = (const v4f*)(x + (size_t)b * Cn * HWn + hw);
    v4f acc = *(const v4f*)(pbsum + phw);
#pragma unroll 8
    for (int c = 0; c < Cn; ++c) {
        v4f xv = xp[(size_t)c * (HWn / 4)];
        v4f wv = *(const v4f*)(pw + c * 256 + phw);
#pragma unroll
        for (int q = 0; q < 4; ++q)
            acc[q] = fmaf(xv[q], wv[q], acc[q]);
    }
    *(v4f*)(S + base) = acc;
}

// ---------------------------------------------------------------------------
// Kernel 2: 3x3 channel-shared conv as fp32 WMMA GEMM.
//   D[o, n] = sum_{k<9} comp[o,k] * S[b, y + k/3, xbase + n + k%3]
// Grid = (OHn, Bn); one wave owns a full output row: loads the 128x12
// zero-padded A (compressor) fragments ONCE, then sweeps the 14 x-blocks,
// each = 6 B loads + 24 v_wmma_f32_16x16x4_f32 + 64 predicated stores.
// ---------------------------------------------------------------------------
__global__ __launch_bounds__(32) void k_conv_wmma(const float* __restrict__ S,
                                                  const float* __restrict__ comp,
                                                  float* __restrict__ out) {
    const int y = blockIdx.x;     // 0..221
    const int b = blockIdx.y;     // 0..15

    const int lane = threadIdx.x; // 0..31
    const int half = lane >> 4;   // 0: lanes 0-15, 1: lanes 16-31
    const int lm   = lane & 15;

    // ---- Load A once: compressor tiles (lane = M%16; VGPR j: K = j+2*half) ----
    v2f A[8][3];
#pragma unroll
    for (int t = 0; t < 8; ++t) {
#pragma unroll
        for (int ks = 0; ks < 3; ++ks) {
#pragma unroll
            for (int j = 0; j < 2; ++j) {
                int k  = 4 * ks + 2 * half + j;
                int kc = k < 9 ? k : 8;                 // clamp: load stays in range
                float v = comp[(t * 16 + lm) * 9 + kc];
                A[t][ks][j] = (k < 9) ? v : 0.0f;       // v_cndmask, EXEC stays full
            }
        }
    }

    // ---- Loop-invariant per-lane B geometry: k -> (kh, kw), valid mask ----
    int boff[3][2];    // row/col offset into S for each (ks, j)
    bool bval[3][2];
#pragma unroll
    for (int ks = 0; ks < 3; ++ks) {
#pragma unroll
        for (int j = 0; j < 2; ++j) {
            int k  = 4 * ks + 2 * half + j;
            int kc = k < 9 ? k : 8;
            boff[ks][j] = (kc / 3) * Wn + (kc % 3);     // kh*224 + kw
            bval[ks][j] = (k < 9);
        }
    }
    const float* srow = S + ((size_t)b * Hn + y) * Wn;   // window row base
    float* orow = out + ((size_t)(b * On + 8 * half) * OHn + y) * OWn + lm;

    for (int xb = 0; xb < NXB; ++xb) {
        const int xbase = xb * 16;
        const int xn = xbase + lm;                       // output column (N)

        // ---- Load B: S windows (lane = N%16; VGPR j: K = j + 2*half) ----
        v2f Bv[3];
#pragma unroll
        for (int ks = 0; ks < 3; ++ks) {
#pragma unroll
            for (int j = 0; j < 2; ++j) {
                int idx = xn + boff[ks][j];
                int lim = (boff[ks][j] / Wn) * Wn + (Wn - 1);
                if (idx > lim) idx = lim;                // clamp padded lanes in-row
                float v = srow[idx];
                Bv[ks][j] = bval[ks][j] ? v : 0.0f;
            }
        }

        // ---- 24 WMMAs: 8 M-tiles x 3 k-steps, fp32 accumulate ----
        v8f acc[8] = {};
#pragma unroll
        for (int ks = 0; ks < 3; ++ks) {
#pragma unroll
            for (int t = 0; t < 8; ++t) {
                acc[t] = __builtin_amdgcn_wmma_f32_16x16x4_f32(
                    /*neg_a=*/false, A[t][ks],
                    /*neg_b=*/false, Bv[ks],
                    /*c_mod=*/(short)0, acc[t],
                    /*reuse_a=*/false, /*reuse_b=*/false);
            }
        }

        // ---- Store: VGPR r -> M = r + 8*half, N = lm ----
        if (xn < OWn) {
#pragma unroll
            for (int t = 0; t < 8; ++t) {
#pragma unroll
                for (int r = 0; r < 8; ++r) {
                    orow[(size_t)(t * 16 + r) * OHWn + xbase] = acc[t][r];
                }
            }
        }
    }
}

// ---------------------------------------------------------------------------
extern "C" void kernel_launch(void* const* d_in, const int* in_sizes, int n_in,
                              void* d_out, int out_size, void* d_ws, size_t ws_size,
                              hipStream_t stream) {
    const float* x    = (const float*)d_in[0];  // (B,C,H,W)
    const float* pw   = (const float*)d_in[1];  // (C,16,16)
    const float* pb   = (const float*)d_in[2];  // (C,16,16)
    const float* comp = (const float*)d_in[3];  // (O,3,3)
    float* out = (float*)d_out;

    float* pbsum = (float*)d_ws;                // 256 floats
    float* S     = pbsum + 256;                 // B*H*W floats (~3.2 MB)

    k_bias_sum<<<1, 256, 0, stream>>>(pb, pbsum);

    const int nvec = Bn * HWn / 4;              // 200704 = 784 * 256 exactly
    k_reduce_c<<<nvec / 256, 256, 0, stream>>>(x, pw, pbsum, S);

    k_conv_wmma<<<dim3(OHn, Bn), 32, 0, stream>>>(S, comp, out);
}